// DiGCN_12859132084303
// MI455X (gfx1250) — compile-verified
//
#include <hip/hip_runtime.h>

typedef __attribute__((ext_vector_type(2))) float v2f;
typedef __attribute__((ext_vector_type(8))) float v8f;

#define NNODES 100000
#define NEDGES 3200000
#define DIN    128
#define DH     64

// ---------------------------------------------------------------------------
// Guaranteed-native f32 atomic add: single L2 atomic, no return, device scope.
// (WGP$ is read-only on CDNA5; RMW executes at the L2 atomic units, the
// correct coherence point for cross-workgroup scatter accumulation.)
// ---------------------------------------------------------------------------
__device__ __forceinline__ void atomic_add_f32(float* p, float v) {
    asm volatile("global_atomic_add_f32 %0, %1, off scope:SCOPE_DEV"
                 :: "v"(p), "v"(v) : "memory");
}

// ---------------------------------------------------------------------------
// C = A[M,K] @ B[K,64] in full fp32 via V_WMMA_F32_16X16X4_F32.
//   * one wave  -> one 16x16 output tile
//   * one block -> 4 waves -> 16 rows x 64 cols
// A layout (ISA 7.12.2, 32-bit A 16x4): lane&15 = M row, half-wave picks the
// K pair -> per-lane A elements are K-contiguous => single global_load_b64.
// B column-per-lane (symmetric to A; ISA: B loaded column-major).
// C/D layout: VGPR j -> M = j + 8*(lane>>4), N = lane&15.
// EXEC all-ones around every WMMA: grid is exact (100000/16 = 6250).
// K is a template parameter so both GEMMs fully unroll (no loop versioning).
// ---------------------------------------------------------------------------
template<int K, bool RELU>
__global__ __launch_bounds__(128)
void gemm_wmma_f32(const float* __restrict__ A, const float* __restrict__ B,
                   float* __restrict__ C) {
    const int lane = threadIdx.x & 31;
    const int wave = threadIdx.x >> 5;     // column tile 0..3
    const int hl   = lane >> 4;            // half-wave group
    const int ml   = lane & 15;
    const int row  = blockIdx.x * 16 + ml; // A row this lane streams
    const int col  = wave * 16 + ml;       // B/C column this lane owns
    const int kg   = hl * 2;               // K offset within a K=4 step

    v8f acc = {};
    const float* __restrict__ arow = A + (size_t)row * K;

#pragma unroll
    for (int kk = 0; kk < K; kk += 4) {
        v2f a = *(const v2f*)(arow + kk + kg);          // global_load_b64
        if (RELU) {
            a.x = __builtin_fmaxf(a.x, 0.0f);
            a.y = __builtin_fmaxf(a.y, 0.0f);
        }
        v2f b;
        b.x = B[(size_t)(kk + kg    ) * DH + col];
        b.y = B[(size_t)(kk + kg + 1) * DH + col];
        // 8 args: (neg_a, A, neg_b, B, c_mod, C, reuse_a, reuse_b)
        acc = __builtin_amdgcn_wmma_f32_16x16x4_f32(
            false, a, false, b, (short)0, acc, false, false);
    }

    float* __restrict__ cbase =
        C + ((size_t)blockIdx.x * 16 + 8 * hl) * DH + col;
#pragma unroll
    for (int j = 0; j < 8; ++j)
        cbase[(size_t)j * DH] = acc[j];
}

// ---------------------------------------------------------------------------
// out[n, f] = bias[f]  (accumulator init so the scatter can pure-add)
// ---------------------------------------------------------------------------
__global__ __launch_bounds__(256)
void init_bias(float* __restrict__ out, const float* __restrict__ bias) {
    const int i = blockIdx.x * 256 + threadIdx.x;
    out[i] = bias[i & (DH - 1)];
}

// ---------------------------------------------------------------------------
// Weighted scatter-add: out[dst] += w * x[src], 64 feats per edge.
// 16 threads per edge, float4 per thread: 128-bit gather + 4 f32 atomics.
// Both x and out are L2-resident (25.6 MB each vs 192 MB L2), so this phase
// is bounded by the L2 atomic-op rate, not HBM.
// ---------------------------------------------------------------------------
__global__ __launch_bounds__(256)
void scatter_edges(const int* __restrict__ ei, const float* __restrict__ ew,
                   const float* __restrict__ x, float* __restrict__ out) {
    const long long t = (long long)blockIdx.x * 256 + threadIdx.x;
    const int e  = (int)(t >> 4);
    const int fg = ((int)t & 15) << 2;

    const int   src = ei[e];
    const int   dst = ei[NEDGES + e];
    const float w   = ew[e];

    const float4 v = *(const float4*)(x + (size_t)src * DH + fg);
    float* o = out + (size_t)dst * DH + fg;
    atomic_add_f32(o + 0, w * v.x);
    atomic_add_f32(o + 1, w * v.y);
    atomic_add_f32(o + 2, w * v.z);
    atomic_add_f32(o + 3, w * v.w);
}

// ---------------------------------------------------------------------------
// inputs: 0 edge_index[2,E] i32 | 1 edge_weight[E] f32 | 2 emb[N,128] f32
//         3 W1[128,64] | 4 b1[64] | 5 W2[64,64] | 6 b2[64]
// output: [N,64] f32
// ---------------------------------------------------------------------------
extern "C" void kernel_launch(void* const* d_in, const int* in_sizes, int n_in,
                              void* d_out, int out_size, void* d_ws, size_t ws_size,
                              hipStream_t stream) {
    const int*   ei  = (const int*)  d_in[0];
    const float* ew  = (const float*)d_in[1];
    const float* emb = (const float*)d_in[2];
    const float* W1  = (const float*)d_in[3];
    const float* b1  = (const float*)d_in[4];
    const float* W2  = (const float*)d_in[5];
    const float* b2  = (const float*)d_in[6];
    float* out = (float*)d_out;

    float* x    = (float*)d_ws;                 // [N,64] dense-transform buf
    float* agg1 = x + (size_t)NNODES * DH;      // [N,64] layer-1 accumulator

    const int gemm_blocks = NNODES / 16;                 // 6250, exact
    const int elem_blocks = (NNODES * DH) / 256;         // 25000, exact
    const int edge_blocks = (int)(((long long)NEDGES * 16) / 256); // 200000, exact

    // Layer 1: x = emb @ W1 ; agg1 = b1 ; agg1 += scatter(w * x[src])
    gemm_wmma_f32<DIN, false><<<gemm_blocks, 128, 0, stream>>>(emb, W1, x);
    init_bias<<<elem_blocks, 256, 0, stream>>>(agg1, b1);
    scatter_edges<<<edge_blocks, 256, 0, stream>>>(ei, ew, x, agg1);

    // Layer 2: x = relu(agg1) @ W2 ; out = b2 ; out += scatter(w * x[src])
    gemm_wmma_f32<DH, true><<<gemm_blocks, 128, 0, stream>>>(agg1, W2, x);
    init_bias<<<elem_blocks, 256, 0, stream>>>(out, b2);
    scatter_edges<<<edge_blocks, 256, 0, stream>>>(ei, ew, x, out);
}